// ForwardBackwardLossBase_49100066128214
// MI455X (gfx1250) — compile-verified
//
#include <hip/hip_runtime.h>
#include <stdint.h>

#define CH        16    // timesteps per double-buffered chunk
#define CELLS     13    // CTC states per lane (13*32 = 416 >= L = 2S+1, S<=207)
#define SLOTS     16    // padded per-lane slot stride in gathered rows (16B-aligned vec loads)
#define ROWW      (32 * SLOTS)  // floats per gathered row = 512
#define VEFF      256   // vocabulary size (reference V)
#define RESCALE   8     // renormalize alpha every 8 timesteps

#if defined(__has_builtin)
# if __has_builtin(__builtin_amdgcn_global_load_async_to_lds_b128)
#  define HAVE_ASYNC_B128 1
# endif
# if __has_builtin(__builtin_amdgcn_s_wait_asynccnt)
#  define HAVE_WAIT_ASYNC 1
# endif
#endif

typedef int v4i __attribute__((ext_vector_type(4)));
typedef v4i g_v4i_t __attribute__((address_space(1)));
typedef v4i l_v4i_t __attribute__((address_space(3)));

__device__ __forceinline__ void async_cp16(void* l, const void* g) {
#ifdef HAVE_ASYNC_B128
  __builtin_amdgcn_global_load_async_to_lds_b128(
      (g_v4i_t*)(unsigned long long)(uintptr_t)g,
      (l_v4i_t*)(unsigned int)(uintptr_t)l,
      0, 0);
#else
  *(float4*)l = *(const float4*)g;  // synchronous fallback
#endif
}

__device__ __forceinline__ void wait_async0() {
#ifdef HAVE_ASYNC_B128
# ifdef HAVE_WAIT_ASYNC
  __builtin_amdgcn_s_wait_asynccnt(0);
# else
  asm volatile("s_wait_asynccnt 0x0" ::: "memory");
# endif
#endif
}

__global__ __launch_bounds__(256, 1)
void ctc_loss_kernel(const float* __restrict__ logits,
                     const int*   __restrict__ targets,
                     const int*   __restrict__ llen,
                     const int*   __restrict__ tlen,
                     float*       __restrict__ out,
                     int T, int S) {
  __shared__ __attribute__((aligned(16))) float raw[2][CH * VEFF];   // raw logit rows
  __shared__ __attribute__((aligned(16))) float gth[2][CH * ROWW];   // gathered probs per state
  __shared__ int   lblL[32 * CELLS];                                 // extended-label table
  __shared__ float afin[32 * CELLS];                                 // final alpha dump

  const int b    = blockIdx.x;
  const int tid  = threadIdx.x;
  const int lane = tid & 31;
  const int wid  = tid >> 5;          // wave 0 = DP wave; waves 1..7 = producers
  const int L    = 2 * S + 1;
  const int Tb   = llen[b];
  const int nch  = (Tb + CH - 1) / CH;
  const int*   tgt = targets + (size_t)b * S;
  const float* lg  = logits  + (size_t)b * T * VEFF;

  // ---- wave0 publishes the extended-label table (blank interleave) ----
  if (wid == 0) {
#pragma unroll
    for (int j = 0; j < CELLS; ++j) {
      int c = lane + 32 * j;
      int v = 0;
      if ((c & 1) && c < L) v = tgt[(c - 1) >> 1];
      lblL[c] = v;
    }
  }
  __syncthreads();

  // ---- DP-wave private state: 13 alpha cells / lane, skip flags ----
  float a[CELLS]     = {};
  float skipf[CELLS] = {};
  float clog = 0.0f;
  int   scnt = 0;
  if (wid == 0) {
#pragma unroll
    for (int i = 0; i < CELLS; ++i) {
      int s = lane * CELLS + i;
      a[i] = (s == 0) ? 1.0f : 0.0f;
      float sk = 0.0f;
      if ((s & 1) && s < L) {
        int k = (s - 1) >> 1;
        sk = (k == 0) ? 1.0f : ((tgt[k] != tgt[k - 1]) ? 1.0f : 0.0f);
      }
      skipf[i] = sk;
    }
  }

  // ---- producer private state: per-cell label + padded slot (constant) ----
  int plbl[CELLS], pslot[CELLS], pvalid[CELLS];
  if (wid > 0) {
#pragma unroll
    for (int j = 0; j < CELLS; ++j) {
      int c = lane + 32 * j;
      plbl[j]   = lblL[c];
      pslot[j]  = (c / CELLS) * SLOTS + (c % CELLS);
      pvalid[j] = (c < L) ? 1 : 0;
    }
  }

  // ---- producer: async-stage a chunk of rows, softmax, gather per-state probs ----
  auto fill = [&](int c, int buf) {
    const int t0 = c * CH;
    // issue all async copies for this wave's rows (rows r = wid-1 + 7q)
#pragma unroll
    for (int q = 0; q < 3; ++q) {
      int r = (wid - 1) + 7 * q;
      if (r < CH && (t0 + r) < Tb) {
        const float* gs = lg + (size_t)(t0 + r) * VEFF + lane * 8;
        float*       ld = &raw[buf][r * VEFF + lane * 8];
        async_cp16(ld,     gs);
        async_cp16(ld + 4, gs + 4);
      }
    }
    wait_async0();
    // per-row: softmax normalizer, then gather probs for the 416 padded states
#pragma unroll
    for (int q = 0; q < 3; ++q) {
      int r = (wid - 1) + 7 * q;
      if (r < CH && (t0 + r) < Tb) {
        const float* rw = &raw[buf][r * VEFF];
        const float4 x0 = *(const float4*)(rw + lane * 8);
        const float4 x1 = *(const float4*)(rw + lane * 8 + 4);
        float mx = fmaxf(fmaxf(fmaxf(x0.x, x0.y), fmaxf(x0.z, x0.w)),
                         fmaxf(fmaxf(x1.x, x1.y), fmaxf(x1.z, x1.w)));
        for (int off = 16; off > 0; off >>= 1) mx = fmaxf(mx, __shfl_xor(mx, off));
        float sm = __expf(x0.x - mx) + __expf(x0.y - mx) + __expf(x0.z - mx) + __expf(x0.w - mx)
                 + __expf(x1.x - mx) + __expf(x1.y - mx) + __expf(x1.z - mx) + __expf(x1.w - mx);
        for (int off = 16; off > 0; off >>= 1) sm += __shfl_xor(sm, off);
        float rZ = 1.0f / sm;
        float* gr = &gth[buf][r * ROWW];
#pragma unroll
        for (int j = 0; j < CELLS; ++j) {
          float v = 0.0f;
          if (pvalid[j]) v = __expf(rw[plbl[j]] - mx) * rZ;
          gr[pslot[j]] = v;
        }
      }
    }
  };

  // ---- DP wave: band-3 recurrence in scaled-probability domain ----
  auto dpchunk = [&](int c, int buf) {
    const int t0   = c * CH;
    const int tmax = (Tb - t0 < CH) ? (Tb - t0) : CH;
    const float* gb = &gth[buf][0];
    for (int tt = 0; tt < tmax; ++tt) {
      const float* row = gb + tt * ROWW + lane * SLOTS;
      const float4 q0 = *(const float4*)(row);
      const float4 q1 = *(const float4*)(row + 4);
      const float4 q2 = *(const float4*)(row + 8);
      float p[CELLS];
      p[0] = q0.x; p[1] = q0.y; p[2]  = q0.z; p[3]  = q0.w;
      p[4] = q1.x; p[5] = q1.y; p[6]  = q1.z; p[7]  = q1.w;
      p[8] = q2.x; p[9] = q2.y; p[10] = q2.z; p[11] = q2.w;
      p[12] = row[12];
      float h1 = __shfl_up(a[12], 1);
      float h2 = __shfl_up(a[11], 1);
      if (lane == 0) { h1 = 0.0f; h2 = 0.0f; }
      float n[CELLS];
      n[0] = (a[0] + h1   + skipf[0] * h2) * p[0];
      n[1] = (a[1] + a[0] + skipf[1] * h1) * p[1];
#pragma unroll
      for (int i = 2; i < CELLS; ++i)
        n[i] = (a[i] + a[i - 1] + skipf[i] * a[i - 2]) * p[i];
#pragma unroll
      for (int i = 0; i < CELLS; ++i) a[i] = n[i];
      if (++scnt == RESCALE) {           // renormalize; accumulate log-scale
        scnt = 0;
        float s = 0.0f;
#pragma unroll
        for (int i = 0; i < CELLS; ++i) s += a[i];
        for (int off = 16; off > 0; off >>= 1) s += __shfl_xor(s, off);
        float ri = 1.0f / s;
        clog += logf(s);
#pragma unroll
        for (int i = 0; i < CELLS; ++i) a[i] *= ri;
      }
    }
  };

  // ---- double-buffered pipeline: producers fill chunk c+1 while wave0 runs chunk c ----
  if (wid > 0) fill(0, 0);
  __syncthreads();
  for (int c = 0; c < nch; ++c) {
    int buf = c & 1;
    if (wid > 0) {
      if (c + 1 < nch) fill(c + 1, buf ^ 1);
    } else {
      dpchunk(c, buf);
    }
    __syncthreads();
  }

  // ---- finalize: loss = -(log(alpha[2l-1] + alpha[2l]) + accumulated log scales) ----
  if (wid == 0) {
#pragma unroll
    for (int i = 0; i < CELLS; ++i) afin[lane * CELLS + i] = a[i];
  }
  __syncthreads();
  if (tid == 0) {
    int tl = tlen[b];
    float g1 = afin[2 * tl - 1];
    float g2 = afin[2 * tl];
    out[b] = -(logf(g1 + g2) + clog);
  }
}

extern "C" void kernel_launch(void* const* d_in, const int* in_sizes, int n_in,
                              void* d_out, int out_size, void* d_ws, size_t ws_size,
                              hipStream_t stream) {
  const float* logits  = (const float*)d_in[0];
  const int*   targets = (const int*)d_in[1];
  const int*   llen    = (const int*)d_in[2];
  const int*   tlen    = (const int*)d_in[3];
  const int B = in_sizes[2];
  const int S = in_sizes[1] / B;
  const int T = in_sizes[0] / (B * VEFF);
  ctc_loss_kernel<<<B, 256, 0, stream>>>(logits, targets, llen, tlen,
                                         (float*)d_out, T, S);
}